// ARAPLoss_7052336300349
// MI455X (gfx1250) — compile-verified
//
#include <hip/hip_runtime.h>
#include <math.h>

// ---------------------------------------------------------------------------
// ARAP edge loss: out[b] = mean_e | ||x[b,k]-x[b,j]||^2 - ||dx[b,k]-dx[b,j]||^2 |
// Gather-bound: everything lives in the 192MB L2; fp32 math is free.
// Wave reduction uses V_WMMA_F32_16X16X4_F32 (A = partials, B = ones).
// ---------------------------------------------------------------------------

typedef __attribute__((ext_vector_type(2))) float v2f;
typedef __attribute__((ext_vector_type(8))) float v8f;

#define TPB 256
#define EPT 4   // edges per thread in the main kernel

// Sum of one float per lane across the full wave32, returned uniform.
// Layouts (CDNA5 ISA 7.12.2):
//   A 16x4 f32: lane L<16 holds A[L][0],A[L][1]; lane L>=16 holds A[L-16][2],A[L-16][3]
//   -> with a.y==0 and B==ones: D[m][n] = p_m + p_{m+16} for all n
//   D 16x16 f32: lane0 VGPRs 0..7 = D[0..7][0]; lane16 VGPRs 0..7 = D[8..15][0]
//   -> total = readlane(sum(c[0..7]), 0) + readlane(sum(c[0..7]), 16)
// Requires EXEC == all ones (call only from fully converged code).
__device__ __forceinline__ float wave_sum32(float v) {
    v2f a;   a[0] = v;    a[1] = 0.0f;
    v2f one; one[0] = 1.0f; one[1] = 1.0f;
    v8f c = {};
    c = __builtin_amdgcn_wmma_f32_16x16x4_f32(
        /*neg_a=*/false, a, /*neg_b=*/false, one,
        /*c_mod=*/(short)0, c, /*reuse_a=*/false, /*reuse_b=*/false);
    float s = ((c[0] + c[1]) + (c[2] + c[3])) + ((c[4] + c[5]) + (c[6] + c[7]));
    int si  = __builtin_bit_cast(int, s);
    float s0  = __builtin_bit_cast(float, __builtin_amdgcn_readlane(si, 0));
    float s16 = __builtin_bit_cast(float, __builtin_amdgcn_readlane(si, 16));
    return s0 + s16;
}

// Deterministic 256-thread block reduction -> single float (valid in thread 0).
__device__ __forceinline__ float block_sum(float v) {
    __shared__ float smem[TPB / 32];
    float ws = wave_sum32(v);                 // uniform per wave
    int lane = threadIdx.x & 31;
    int wid  = threadIdx.x >> 5;
    if (lane == 0) smem[wid] = ws;
    __syncthreads();
    float tot = 0.0f;
    if (threadIdx.x == 0) {
#pragma unroll
        for (int i = 0; i < TPB / 32; ++i) tot += smem[i];
    }
    return tot;
}

__global__ __launch_bounds__(TPB)
void arap_edge_kernel(const float* __restrict__ dx, const float* __restrict__ x,
                      const int* __restrict__ edges, float* __restrict__ partial,
                      int E, int NV) {
    const int b = blockIdx.y;
    const size_t base = (size_t)b * (size_t)NV * 3u;
    const float* __restrict__ xb  = x  + base;
    const float* __restrict__ dxb = dx + base;

    float acc = 0.0f;
    const int t0 = blockIdx.x * (TPB * EPT) + threadIdx.x;
#pragma unroll
    for (int e = 0; e < EPT; ++e) {
        const int t = t0 + e * TPB;          // coalesced edge-index reads
        if (t < E) {
            const int j = edges[2 * t + 0];
            const int k = edges[2 * t + 1];
            const float* xj = xb  + 3 * (size_t)j;
            const float* xk = xb  + 3 * (size_t)k;
            const float* dj = dxb + 3 * (size_t)j;
            const float* dk = dxb + 3 * (size_t)k;
            const float ax = xk[0] - xj[0];
            const float ay = xk[1] - xj[1];
            const float az = xk[2] - xj[2];
            const float bx = dk[0] - dj[0];
            const float by = dk[1] - dj[1];
            const float bz = dk[2] - dj[2];
            const float s1 = ax * ax + ay * ay + az * az;
            const float s2 = bx * bx + by * by + bz * bz;
            acc += fabsf(s1 - s2);
        }
    }
    // all lanes reconverged here -> EXEC all ones for WMMA
    const float tot = block_sum(acc);
    if (threadIdx.x == 0)
        partial[(size_t)b * gridDim.x + blockIdx.x] = tot;
}

__global__ __launch_bounds__(TPB)
void arap_finalize_kernel(const float* __restrict__ partial, float* __restrict__ out,
                          int nblocks, float invE) {
    const int b = blockIdx.x;
    const float* __restrict__ p = partial + (size_t)b * (size_t)nblocks;
    float acc = 0.0f;
    for (int i = threadIdx.x; i < nblocks; i += TPB)  // trip count varies per thread; reconverges before block_sum
        acc += p[i];
    const float tot = block_sum(acc);
    if (threadIdx.x == 0)
        out[b] = tot * invE;
}

extern "C" void kernel_launch(void* const* d_in, const int* in_sizes, int n_in,
                              void* d_out, int out_size, void* d_ws, size_t ws_size,
                              hipStream_t stream) {
    // setup_inputs order: dx (B*NV*3 f32), x (B*NV*3 f32), edges (E*2 i32)
    const float* dxp   = (const float*)d_in[0];
    const float* xp    = (const float*)d_in[1];
    const int*   edges = (const int*)d_in[2];
    float*       out   = (float*)d_out;

    const int B  = out_size;                 // 16
    const int E  = in_sizes[2] / 2;          // ~1.2M unique directed edges
    const int NV = in_sizes[0] / (B * 3);    // 100000

    const int nblocksX = (E + TPB * EPT - 1) / (TPB * EPT);
    float* partial = (float*)d_ws;           // B * nblocksX floats (~75 KB)

    dim3 grid1(nblocksX, B);
    arap_edge_kernel<<<grid1, dim3(TPB), 0, stream>>>(dxp, xp, edges, partial, E, NV);
    arap_finalize_kernel<<<dim3(B), dim3(TPB), 0, stream>>>(partial, out, nblocksX,
                                                            1.0f / (float)E);
}